// DecoderStepStaticWrapper_2473901162610
// MI455X (gfx1250) — compile-verified
//
#include <hip/hip_runtime.h>
#include <hip/hip_bf16.h>

// ---------------- problem constants (from reference) ----------------
#define L_      2
#define D_      2048
#define HQ_     16
#define HKV_    8
#define HD_     128
#define FF_     8192
#define V_      32000
#define MAXLEN_ 4096
#define NREP_   (HQ_ / HKV_)
#define EPS_    1e-6f
#define SCALING_ 0.08838834764831845f   // 128^-0.5
#define THETA_  1000000.0f

typedef float v2f __attribute__((ext_vector_type(2)));
typedef float v8f __attribute__((ext_vector_type(8)));
typedef unsigned int u32x4 __attribute__((ext_vector_type(4)));
typedef int i32x4 __attribute__((ext_vector_type(4)));
typedef int i32x8 __attribute__((ext_vector_type(8)));

// ---------------- reduction helpers (wave32) ----------------
__device__ inline float wave_sum(float v) {
    #pragma unroll
    for (int o = 16; o > 0; o >>= 1) v += __shfl_xor(v, o, 32);
    return v;
}
__device__ inline float wave_max(float v) {
    #pragma unroll
    for (int o = 16; o > 0; o >>= 1) v = fmaxf(v, __shfl_xor(v, o, 32));
    return v;
}
__device__ inline float block_sum(float v, float* red) {
    int lane = threadIdx.x & 31, wid = threadIdx.x >> 5, nw = blockDim.x >> 5;
    v = wave_sum(v);
    if (lane == 0) red[wid] = v;
    __syncthreads();
    float r = (lane < nw) ? red[lane] : 0.0f;
    r = wave_sum(r);
    __syncthreads();
    return r;
}
__device__ inline float block_max(float v, float* red) {
    int lane = threadIdx.x & 31, wid = threadIdx.x >> 5, nw = blockDim.x >> 5;
    v = wave_max(v);
    if (lane == 0) red[wid] = v;
    __syncthreads();
    float r = (lane < nw) ? red[lane] : -3.0e38f;
    r = wave_max(r);
    __syncthreads();
    return r;
}

// ---------------- simple elementwise kernels ----------------
__global__ void copy_kernel(const float* __restrict__ src, float* __restrict__ dst, int n) {
    int i = blockIdx.x * blockDim.x + threadIdx.x;
    if (i < n) dst[i] = src[i];
}

__global__ void silu_mul_kernel(const float* __restrict__ g, const float* __restrict__ u,
                                float* __restrict__ out, int n) {
    int i = blockIdx.x * blockDim.x + threadIdx.x;
    if (i < n) {
        float gv = g[i];
        out[i] = gv * (1.0f / (1.0f + __expf(-gv))) * u[i];
    }
}

// ---------------- RMSNorm: one block, generic length ----------------
__global__ void rmsnorm_kernel(const float* __restrict__ x, const float* __restrict__ w,
                               float* __restrict__ out, int n) {
    __shared__ float red[8];
    float s = 0.0f;
    for (int i = threadIdx.x; i < n; i += blockDim.x) { float v = x[i]; s += v * v; }
    float tot = block_sum(s, red);
    float inv = rsqrtf(tot / (float)n + EPS_);
    for (int i = threadIdx.x; i < n; i += blockDim.x) out[i] = x[i] * inv * w[i];
}

// ---------------- WMMA fp32 matvec: y[n] = sum_k x[k]*W[k*N+n] (+base) ----
__global__ void matvec_wmma_kernel(const float* __restrict__ x,
                                   const float* __restrict__ W,
                                   const float* __restrict__ base,   // may be null
                                   float* __restrict__ y,
                                   int K, int N) {
    int wave = (int)((blockIdx.x * blockDim.x + threadIdx.x) >> 5);
    int lane = threadIdx.x & 31;
    int n0 = wave * 16;
    if (n0 >= N) return;                 // wave-uniform: EXEC stays all-ones
    int half = lane >> 4;                // 0: K{0,1}   1: K{2,3}
    int col  = lane & 15;                // row index for A, column index for B
    v8f c = {};
    for (int k = 0; k < K; k += 4) {
        float x0 = x[k + half * 2 + 0];
        float x1 = x[k + half * 2 + 1];
        v2f a, b;
        a.x = (col == 0) ? x0 : 0.0f;    // only row M=0 of A is live
        a.y = (col == 0) ? x1 : 0.0f;
        const float* wr = W + (size_t)k * N + n0 + col;
        b.x = wr[(size_t)(half * 2 + 0) * N];
        b.y = wr[(size_t)(half * 2 + 1) * N];
        __builtin_prefetch(wr + 8 * (size_t)N, 0, 1);
        c = __builtin_amdgcn_wmma_f32_16x16x4_f32(false, a, false, b,
                                                  (short)0, c, false, false);
    }
    if (lane < 16) {
        int n = n0 + lane;
        if (n < N) {
            float r = c[0];              // C: VGPR0, lanes 0..15 = row M=0
            y[n] = base ? (base[n] + r) : r;
        }
    }
}

// ---------------- fused per-head RMSNorm + RoPE for q and k -----------
__global__ void qknorm_rope_kernel(float* __restrict__ q, float* __restrict__ k,
                                   const float* __restrict__ qn,
                                   const float* __restrict__ kn,
                                   const long long* __restrict__ pos_ids) {
    __shared__ float buf[HD_];
    __shared__ float red[4];
    int h = blockIdx.x;
    float* x; const float* w;
    if (h < HQ_) { x = q + (size_t)h * HD_;          w = qn; }
    else         { x = k + (size_t)(h - HQ_) * HD_;  w = kn; }
    int t = threadIdx.x;
    float v = x[t];
    float tot = block_sum(v * v, red);
    float inv = rsqrtf(tot / (float)HD_ + EPS_);
    float nv = v * inv * w[t];
    buf[t] = nv;
    __syncthreads();
    float pos = (float)pos_ids[0];
    int j = t & 63;                                       // freq index (d % 64)
    float inv_freq = __powf(THETA_, -((float)(2 * j) / (float)HD_));
    float ang = pos * inv_freq;
    float cs = __cosf(ang), sn = __sinf(ang);
    float rot = (t < 64) ? -buf[t + 64] : buf[t - 64];    // rotate_half
    x[t] = nv * cs + rot * sn;
}

// ---------------- TDM helper: 1D tensor load (global -> LDS) ----------
// Builds a D# for a 1x nelts fp32 tile and issues TENSOR_LOAD_TO_LDS.
// This toolchain carries the 6-arg builtin:
//   (u32x4 g0, i32x8 g1, i32x4 g2, i32x4 g3, i32x8 g4, i32 cpol)
__device__ inline void tdm_load_1d(const float* gsrc, unsigned lds_byte_off, int nelts) {
    unsigned long long ga = (unsigned long long)gsrc;
    u32x4 g0;
    g0.x = 0x1u;                                          // count=1, user descriptor
    g0.y = lds_byte_off;                                  // lds_addr
    g0.z = (unsigned)(ga & 0xFFFFFFFFull);                // global_addr[31:0]
    g0.w = (unsigned)((ga >> 32) & 0x01FFFFFFull)         // global_addr[56:32]
         | (2u << 30);                                    // type = 2 ("image")
    i32x8 g1;
    g1[0] = (2 << 16);                                    // data_size=2 (4B), wg_mask=0
    g1[1] = (nelts & 0xFFFF) << 16;                       // tensor_dim0 [15:0] -> bits[63:48]
    g1[2] = ((nelts >> 16) & 0xFFFF) | (1 << 16);         // dim0 hi | tensor_dim1=1
    g1[3] = (nelts & 0xFFFF) << 16;                       // tile_dim0 -> bits[127:112]
    g1[4] = 1;                                            // tile_dim1=1 -> bits[143:128]
    g1[5] = nelts;                                        // tensor_dim0_stride low 32
    g1[6] = 0;
    g1[7] = 0;
    i32x4 z4 = {0, 0, 0, 0};
    i32x8 z8 = {0, 0, 0, 0, 0, 0, 0, 0};
    __builtin_amdgcn_tensor_load_to_lds(g0, g1, z4, z4, z8, 0);
}

// ---------------- KV-cache update staged through LDS via TDM ----------
__global__ void writekv_tdm_kernel(const float* __restrict__ kin,
                                   const float* __restrict__ vin,
                                   float* __restrict__ kout,
                                   float* __restrict__ vout,
                                   const int* __restrict__ kvp) {
    __shared__ float stage[2 * HKV_ * HD_];               // 8 KB at LDS base
    int t = threadIdx.x;
    if (t < 32) {                                         // wave 0 issues DMA
        tdm_load_1d(kin, 0, HKV_ * HD_);
        tdm_load_1d(vin, HKV_ * HD_ * 4, HKV_ * HD_);
        __builtin_amdgcn_s_wait_tensorcnt(0);
    }
    __syncthreads();
    int kv_pos = *kvp;
    for (int i = t; i < HKV_ * HD_; i += blockDim.x) {
        int h = i / HD_, d = i % HD_;
        size_t off = (size_t)h * MAXLEN_ * HD_ + (size_t)kv_pos * HD_ + d;
        kout[off] = stage[i];
        vout[off] = stage[HKV_ * HD_ + i];
    }
}

// ---------------- attention: one block per q head ---------------------
// Phase 3 streams V through LDS with async global->LDS copies (ASYNCcnt).
#define CH_ 64                                            // V rows per LDS chunk
__global__ void attention_kernel(const float* __restrict__ q,
                                 const float* __restrict__ Kc,
                                 const float* __restrict__ Vc,
                                 float* __restrict__ scores,
                                 float* __restrict__ out,
                                 const int* __restrict__ kvp) {
    __shared__ float vbuf[CH_ * HD_];                     // 32 KB, at LDS base
    __shared__ float qs[HD_];
    __shared__ float red[8];
    __shared__ float sh_stat[2];
    int head = blockIdx.x;
    int kvh  = head / NREP_;
    int alen = *kvp + 1;
    const float* qh = q + (size_t)head * HD_;
    const float* Kr = Kc + (size_t)kvh * MAXLEN_ * HD_;
    const float* Vr = Vc + (size_t)kvh * MAXLEN_ * HD_;
    float* sc = scores + (size_t)head * MAXLEN_;
    int t = threadIdx.x;
    if (t < HD_) qs[t] = qh[t];
    __syncthreads();

    // phase 1: scores = q . K^T * scale
    float lmax = -3.0e38f;
    for (int p = t; p < alen; p += blockDim.x) {
        const float4* kr = (const float4*)(Kr + (size_t)p * HD_);
        float acc = 0.0f;
        #pragma unroll 8
        for (int d = 0; d < HD_ / 4; ++d) {
            float4 kv = kr[d];
            acc += kv.x * qs[4*d+0] + kv.y * qs[4*d+1]
                 + kv.z * qs[4*d+2] + kv.w * qs[4*d+3];
        }
        acc *= SCALING_;
        sc[p] = acc;
        lmax = fmaxf(lmax, acc);
    }
    float m = block_max(lmax, red);
    if (t == 0) sh_stat[0] = m;
    __syncthreads();
    m = sh_stat[0];

    // phase 2: exp + sum
    float lsum = 0.0f;
    for (int p = t; p < alen; p += blockDim.x) {
        float e = __expf(sc[p] - m);
        sc[p] = e;
        lsum += e;
    }
    float ssum = block_sum(lsum, red);
    if (t == 0) sh_stat[1] = ssum;
    __syncthreads();
    ssum = sh_stat[1];

    // phase 3: attn @ V, V staged through LDS with async copies
    float acc = 0.0f;
    for (int p0 = 0; p0 < alen; p0 += CH_) {
        int rows = min(CH_, alen - p0);
        int nvec = rows * (HD_ / 4);                      // float4 transfers
        for (int idx = t; idx < nvec; idx += blockDim.x) {
            const float* gsrc = Vr + (size_t)p0 * HD_ + (size_t)idx * 4;
            unsigned lds_off = (unsigned)(unsigned long long)(&vbuf[idx * 4]);
            asm volatile("global_load_async_to_lds_b128 %0, %1, off"
                         :: "v"(lds_off), "v"(gsrc) : "memory");
        }
        asm volatile("s_wait_asynccnt 0" ::: "memory");
        __syncthreads();
        if (t < HD_) {
            for (int r = 0; r < rows; ++r)
                acc += sc[p0 + r] * vbuf[r * HD_ + t];
        }
        __syncthreads();
    }
    if (t < HD_) out[(size_t)head * HD_ + t] = acc / ssum;
}

// ---------------- host orchestration ----------------------------------
static inline void launch_matvec(const float* x, const float* W, const float* base,
                                 float* y, int K, int N, hipStream_t s) {
    int grid = (N + 127) / 128;      // 8 waves/block * 16 cols/wave
    matvec_wmma_kernel<<<grid, 256, 0, s>>>(x, W, base, y, K, N);
}

extern "C" void kernel_launch(void* const* d_in, const int* in_sizes, int n_in,
                              void* d_out, int out_size, void* d_ws, size_t ws_size,
                              hipStream_t stream) {
    const float* input_embeds = (const float*)d_in[0];
    const float* past_keys    = (const float*)d_in[1];
    const float* past_values  = (const float*)d_in[2];
    const float* ln1_w        = (const float*)d_in[3];
    const float* q_w          = (const float*)d_in[4];
    const float* k_w          = (const float*)d_in[5];
    const float* v_w          = (const float*)d_in[6];
    const float* o_w          = (const float*)d_in[7];
    const float* qn_w         = (const float*)d_in[8];
    const float* kn_w         = (const float*)d_in[9];
    const float* ln2_w        = (const float*)d_in[10];
    const float* gate_w       = (const float*)d_in[11];
    const float* up_w         = (const float*)d_in[12];
    const float* down_w       = (const float*)d_in[13];
    const float* norm_w       = (const float*)d_in[14];
    const float* lm_head_w    = (const float*)d_in[15];
    const long long* pos_ids  = (const long long*)d_in[16];
    const int* kvp            = (const int*)d_in[17];

    // d_out = logits | present_keys | present_values
    float* logits = (float*)d_out;
    float* pk_out = logits + V_;
    float* pv_out = pk_out + (size_t)L_ * HKV_ * MAXLEN_ * HD_;

    size_t cache_bytes = (size_t)L_ * HKV_ * MAXLEN_ * HD_ * sizeof(float);
    (void)hipMemcpyAsync(pk_out, past_keys,   cache_bytes, hipMemcpyDeviceToDevice, stream);
    (void)hipMemcpyAsync(pv_out, past_values, cache_bytes, hipMemcpyDeviceToDevice, stream);

    // workspace carve-up
    float* ws   = (float*)d_ws;
    float* h    = ws;  ws += D_;
    float* nrm  = ws;  ws += D_;
    float* qb   = ws;  ws += HQ_ * HD_;
    float* kb   = ws;  ws += HKV_ * HD_;
    float* vb   = ws;  ws += HKV_ * HD_;
    float* attn = ws;  ws += HQ_ * HD_;
    float* gate = ws;  ws += FF_;
    float* up   = ws;  ws += FF_;
    float* act  = ws;  ws += FF_;
    float* sc   = ws;  ws += (size_t)HQ_ * MAXLEN_;

    copy_kernel<<<(D_ + 255) / 256, 256, 0, stream>>>(input_embeds, h, D_);

    for (int l = 0; l < L_; ++l) {
        rmsnorm_kernel<<<1, 256, 0, stream>>>(h, ln1_w + (size_t)l * D_, nrm, D_);

        launch_matvec(nrm, q_w + (size_t)l * D_ * (HQ_ * HD_), nullptr, qb, D_, HQ_ * HD_, stream);
        launch_matvec(nrm, k_w + (size_t)l * D_ * (HKV_ * HD_), nullptr, kb, D_, HKV_ * HD_, stream);
        launch_matvec(nrm, v_w + (size_t)l * D_ * (HKV_ * HD_), nullptr, vb, D_, HKV_ * HD_, stream);

        qknorm_rope_kernel<<<HQ_ + HKV_, 128, 0, stream>>>(
            qb, kb, qn_w + (size_t)l * HD_, kn_w + (size_t)l * HD_, pos_ids);

        float* Kc = pk_out + (size_t)l * HKV_ * MAXLEN_ * HD_;
        float* Vc = pv_out + (size_t)l * HKV_ * MAXLEN_ * HD_;
        writekv_tdm_kernel<<<1, 256, 0, stream>>>(kb, vb, Kc, Vc, kvp);

        attention_kernel<<<HQ_, 256, 0, stream>>>(qb, Kc, Vc, sc, attn, kvp);

        // o-proj with fused residual add into h
        launch_matvec(attn, o_w + (size_t)l * (HQ_ * HD_) * D_, h, h, HQ_ * HD_, D_, stream);

        rmsnorm_kernel<<<1, 256, 0, stream>>>(h, ln2_w + (size_t)l * D_, nrm, D_);
        launch_matvec(nrm, gate_w + (size_t)l * D_ * FF_, nullptr, gate, D_, FF_, stream);
        launch_matvec(nrm, up_w   + (size_t)l * D_ * FF_, nullptr, up,   D_, FF_, stream);
        silu_mul_kernel<<<(FF_ + 255) / 256, 256, 0, stream>>>(gate, up, act, FF_);
        // down-proj with fused residual add into h
        launch_matvec(act, down_w + (size_t)l * FF_ * D_, h, h, FF_, D_, stream);
    }

    rmsnorm_kernel<<<1, 256, 0, stream>>>(h, norm_w, nrm, D_);
    launch_matvec(nrm, lm_head_w, nullptr, logits, D_, V_, stream);
}